// PointNetFeatures_18794776887989
// MI455X (gfx1250) — compile-verified
//
#include <hip/hip_runtime.h>
#include <hip/hip_bf16.h>

#define B_    64
#define N_    1024
#define KNN   20
#define CH    64
#define EPS   1e-5f

typedef __attribute__((ext_vector_type(2))) float v2f;
typedef __attribute__((ext_vector_type(8))) float v8f;

// ---------------------------------------------------------------------------
// Kernel 1: kNN.  Each wave owns 16 query rows; distance-key tiles (16x16)
// are produced by V_WMMA_F32_16X16X4_F32 with A=[x0,x1,x2,-1], B=[2x0,2x1,2x2,xx_j]
// so key = 2*gram - xx_j (row-monotonic equivalent of pd).  Tiles bounce
// through LDS; lanes 0-15 each keep a sorted top-20 (val,idx) in registers.
// ---------------------------------------------------------------------------
__global__ __launch_bounds__(256) void knn_topk_kernel(const float* __restrict__ x,
                                                       int* __restrict__ idx_out)
{
    __shared__ float tile[8][16][17];                 // [wave][row][col], padded
    const int lane = threadIdx.x & 31;
    const int wave = threadIdx.x >> 5;
    const int tiles_per_b = N_ / 128;                 // 8 blocks-worth of rows per batch
    const int b    = blockIdx.x / tiles_per_b;
    const int row0 = (blockIdx.x % tiles_per_b) * 128 + wave * 16;
    const float* xb = x + (size_t)b * 3 * N_;

    const int m = lane & 15;
    const int i = row0 + m;
    const float xi0 = xb[i], xi1 = xb[N_ + i], xi2 = xb[2 * N_ + i];
    // A operand (16x4 f32): lanes 0-15 hold K=0,1; lanes 16-31 hold K=2,3
    v2f A;
    A.x = (lane < 16) ? xi0 : xi2;
    A.y = (lane < 16) ? xi1 : -1.0f;

    float bestv[KNN];
    int   bestj[KNN];
#pragma unroll
    for (int t = 0; t < KNN; ++t) { bestv[t] = -3.0e38f; bestj[t] = 0; }

    for (int jt = 0; jt < N_ / 16; ++jt) {
        const int j = jt * 16 + m;
        const float xj0 = xb[j], xj1 = xb[N_ + j], xj2 = xb[2 * N_ + j];
        const float xxj = xj0 * xj0 + xj1 * xj1 + xj2 * xj2;
        // B operand (4x16 f32): lanes 0-15 rows K=0,1; lanes 16-31 rows K=2,3
        v2f Bop;
        Bop.x = (lane < 16) ? (2.0f * xj0) : (2.0f * xj2);
        Bop.y = (lane < 16) ? (2.0f * xj1) : xxj;

        v8f acc = {};
        acc = __builtin_amdgcn_wmma_f32_16x16x4_f32(false, A, false, Bop,
                                                    (short)0, acc, false, false);
        // C/D layout: VGPR c -> row c (lanes 0-15) / row 8+c (lanes 16-31)
#pragma unroll
        for (int c = 0; c < 8; ++c) {
            const int r = (lane < 16) ? c : (8 + c);
            tile[wave][r][m] = acc[c];
        }
        __syncthreads();
        if (lane < 16) {
#pragma unroll
            for (int c2 = 0; c2 < 16; ++c2) {
                float vpd = tile[wave][lane][c2];
                int   jj  = jt * 16 + c2;
                if (vpd > bestv[KNN - 1]) {           // cheap reject path
#pragma unroll
                    for (int t = 0; t < KNN; ++t) {   // register bubble-insert
                        const bool sw = vpd > bestv[t];
                        const float tv = bestv[t]; const int tj = bestj[t];
                        bestv[t] = sw ? vpd : tv;  bestj[t] = sw ? jj : tj;
                        vpd      = sw ? tv : vpd;  jj       = sw ? tj : jj;
                    }
                }
            }
        }
        __syncthreads();
    }
    if (lane < 16) {
        int* o = idx_out + ((size_t)b * N_ + (row0 + lane)) * KNN;
#pragma unroll
        for (int t = 0; t < KNN; ++t) o[t] = bestj[t];
    }
}

// ---------------------------------------------------------------------------
// Kernel 2: LAB attention.  One wave per point.  feat (10x20, padded 12x32)
// built in LDS; v = lrelu(bn(Wv @ feat)) via 4x2 output tiles of chained
// 16x16x4 f32 WMMAs (K padded 10->12 with zero rows); then s/softmax/agg
// stay inside the wave's LDS slice.
// ---------------------------------------------------------------------------
#define WAVES2 4
__global__ __launch_bounds__(128) void lab_attn_kernel(
        const float* __restrict__ x, const int* __restrict__ knn_idx,
        const float* __restrict__ Wk, const float* __restrict__ Wv,
        const float* __restrict__ Ws, const float* __restrict__ bnk,
        const float* __restrict__ bnv, const float* __restrict__ bns,
        float* __restrict__ agg_out)
{
    __shared__ float feat_sh[WAVES2][12][33];
    __shared__ float v_sh[WAVES2][CH][33];
    __shared__ float q_sh[WAVES2][CH];
    __shared__ float sc_sh[WAVES2][32];

    const int lane = threadIdx.x & 31;
    const int w    = threadIdx.x >> 5;
    const int p    = blockIdx.x * WAVES2 + w;         // global point id
    const int b    = p >> 10;
    const int n    = p & (N_ - 1);
    const float* xb = x + (size_t)b * 3 * N_;

    const float c0 = xb[n], c1 = xb[N_ + n], c2 = xb[2 * N_ + n];

    // ---- gather neighbors, build padded feat in LDS (col = lane) ----
    {
        const int col = lane;
        float f[10];
#pragma unroll
        for (int r = 0; r < 10; ++r) f[r] = 0.0f;
        if (col < KNN) {
            const int j  = knn_idx[(size_t)p * KNN + col];
            const float n0 = xb[j], n1 = xb[N_ + j], n2 = xb[2 * N_ + j];
            const float d0 = n0 - c0, d1 = n1 - c1, d2 = n2 - c2;
            f[0] = d0 * d0 + d1 * d1 + d2 * d2;
            f[1] = d0; f[2] = d1; f[3] = d2;
            f[4] = c0; f[5] = c1; f[6] = c2;
            f[7] = n0; f[8] = n1; f[9] = n2;
        }
#pragma unroll
        for (int r = 0; r < 10; ++r) feat_sh[w][r][col] = f[r];
        feat_sh[w][10][col] = 0.0f;
        feat_sh[w][11][col] = 0.0f;
    }
    // ---- q = lrelu(bn(Wk @ x_n)), 2 channels per lane ----
#pragma unroll
    for (int t = 0; t < 2; ++t) {
        const int ch = lane + 32 * t;
        float z = Wk[ch * 3 + 0] * c0 + Wk[ch * 3 + 1] * c1 + Wk[ch * 3 + 2] * c2;
        z = (z - bnk[128 + ch]) * bnk[ch] * rsqrtf(bnk[192 + ch] + EPS) + bnk[64 + ch];
        q_sh[w][ch] = (z >= 0.0f) ? z : 0.2f * z;
    }
    __syncthreads();

    // ---- v = lrelu(bn(Wv @ feat)) via WMMA ----
    const int halfk = (lane < 16) ? 0 : 2;
#pragma unroll
    for (int mt = 0; mt < 4; ++mt) {
#pragma unroll
        for (int nt = 0; nt < 2; ++nt) {
            const int col = (lane & 15) + nt * 16;
            v8f acc = {};
#pragma unroll
            for (int ks = 0; ks < 3; ++ks) {
                const int ka   = ks * 4 + halfk;
                const int mrow = mt * 16 + (lane & 15);
                v2f Aop, Bop;
                Aop.x = (ka     < 10) ? Wv[mrow * 10 + ka]     : 0.0f;
                Aop.y = (ka + 1 < 10) ? Wv[mrow * 10 + ka + 1] : 0.0f;
                Bop.x = feat_sh[w][ka][col];
                Bop.y = feat_sh[w][ka + 1][col];
                acc = __builtin_amdgcn_wmma_f32_16x16x4_f32(false, Aop, false, Bop,
                                                            (short)0, acc, false, false);
            }
#pragma unroll
            for (int c = 0; c < 8; ++c) {
                const int ch = mt * 16 + ((lane < 16) ? c : 8 + c);
                float z = (acc[c] - bnv[128 + ch]) * bnv[ch] *
                          rsqrtf(bnv[192 + ch] + EPS) + bnv[64 + ch];
                v_sh[w][ch][col] = (z >= 0.0f) ? z : 0.2f * z;
            }
        }
    }
    __syncthreads();

    // ---- s = lrelu(bn(Ws @ (q+v))) ----
    {
        const int col = lane;
        float sval = 0.0f;
        if (col < KNN) {
            float a = 0.0f;
#pragma unroll
            for (int ch = 0; ch < CH; ++ch)
                a += Ws[ch] * (q_sh[w][ch] + v_sh[w][ch][col]);
            a = (a - bns[2]) * bns[0] * rsqrtf(bns[3] + EPS) + bns[1];
            sval = (a >= 0.0f) ? a : 0.2f * a;
        }
        sc_sh[w][col] = sval;
    }
    __syncthreads();
    // ---- softmax over the 20 neighbors ----
    {
        const int col = lane;
        float pval = 0.0f;
        if (col < KNN) {
            float mx = -3.0e38f;
#pragma unroll
            for (int k = 0; k < KNN; ++k) mx = fmaxf(mx, sc_sh[w][k]);
            float sm = 0.0f;
#pragma unroll
            for (int k = 0; k < KNN; ++k) sm += __expf(sc_sh[w][k] - mx);
            pval = __expf(sc_sh[w][col] - mx) / sm;
        }
        __syncthreads();
        sc_sh[w][col] = pval;
    }
    __syncthreads();
    // ---- agg[c] = sum_k v[c,k] * p[k] ----
#pragma unroll
    for (int t = 0; t < 2; ++t) {
        const int ch = lane + 32 * t;
        float a = 0.0f;
#pragma unroll
        for (int k = 0; k < KNN; ++k) a += v_sh[w][ch][k] * sc_sh[w][k];
        agg_out[((size_t)b * CH + ch) * N_ + n] = a;
    }
}

// ---------------------------------------------------------------------------
// Kernel 3: conv2 (64x64 GEMM over 65536 columns) + BN + ReLU.
// One wave = one 16(out)x16(col) tile, 16 chained 16x16x4 f32 WMMAs.
// Block = 8 waves = full 64 channels x 32 columns.
// ---------------------------------------------------------------------------
__global__ __launch_bounds__(256) void conv2_kernel(
        const float* __restrict__ agg, const float* __restrict__ W2,
        const float* __restrict__ b2, const float* __restrict__ bnc,
        float* __restrict__ out)
{
    const int lane = threadIdx.x & 31;
    const int w    = threadIdx.x >> 5;
    const int mt   = w & 3;
    const int nt   = w >> 2;
    const int colg = blockIdx.x * 32 + nt * 16 + (lane & 15);
    const int b    = colg >> 10;
    const int n    = colg & (N_ - 1);
    const float* ab = agg + (size_t)b * CH * N_;
    const int halfk = (lane < 16) ? 0 : 2;

    v8f acc = {};
#pragma unroll
    for (int ks = 0; ks < 16; ++ks) {
        const int k0   = ks * 4 + halfk;
        const int mrow = mt * 16 + (lane & 15);
        v2f Aop, Bop;
        Aop.x = W2[mrow * CH + k0];
        Aop.y = W2[mrow * CH + k0 + 1];
        Bop.x = ab[(size_t)k0 * N_ + n];
        Bop.y = ab[(size_t)(k0 + 1) * N_ + n];
        acc = __builtin_amdgcn_wmma_f32_16x16x4_f32(false, Aop, false, Bop,
                                                    (short)0, acc, false, false);
    }
#pragma unroll
    for (int c = 0; c < 8; ++c) {
        const int ch = mt * 16 + ((lane < 16) ? c : 8 + c);
        float z = acc[c] + b2[ch];
        z = (z - bnc[128 + ch]) * bnc[ch] * rsqrtf(bnc[192 + ch] + EPS) + bnc[64 + ch];
        out[((size_t)b * CH + ch) * N_ + n] = fmaxf(z, 0.0f);
    }
}

// ---------------------------------------------------------------------------
extern "C" void kernel_launch(void* const* d_in, const int* in_sizes, int n_in,
                              void* d_out, int out_size, void* d_ws, size_t ws_size,
                              hipStream_t stream)
{
    const float* x   = (const float*)d_in[0];
    const float* Wk  = (const float*)d_in[1];
    const float* Wv  = (const float*)d_in[2];
    const float* Ws  = (const float*)d_in[3];
    const float* W2  = (const float*)d_in[4];
    const float* b2  = (const float*)d_in[5];
    const float* bnk = (const float*)d_in[6];
    const float* bnv = (const float*)d_in[7];
    const float* bns = (const float*)d_in[8];
    const float* bnc = (const float*)d_in[9];

    int*   idxws = (int*)d_ws;                                    // [B,N,KNN]   5.2 MB
    float* agg   = (float*)((char*)d_ws +
                            (size_t)B_ * N_ * KNN * sizeof(int)); // [B,64,N]   16.8 MB

    knn_topk_kernel<<<B_ * (N_ / 128), 256, 0, stream>>>(x, idxws);
    lab_attn_kernel<<<(B_ * N_) / WAVES2, 128, 0, stream>>>(x, idxws, Wk, Wv, Ws,
                                                            bnk, bnv, bns, agg);
    conv2_kernel<<<(B_ * N_) / 32, 256, 0, stream>>>(agg, W2, b2, bnc, (float*)d_out);
}